// neural_rendering_gbuffer_31421980737550
// MI455X (gfx1250) — compile-verified
//
#include <hip/hip_runtime.h>
#include <math.h>

#define IMG_H 240
#define IMG_W 320

typedef __attribute__((ext_vector_type(2))) float v2f;
typedef __attribute__((ext_vector_type(8))) float v8f;

struct ProjMat { float p[16]; };   // P row-major: P[n][k] = p[n*4+k]; clip[n] = sum_k cam[k]*P[n][k]

// ---------------- Kernel 1: face normals (normalized) ----------------
__global__ void k_face_normals(const float* __restrict__ verts,
                               const int* __restrict__ faces,
                               float* __restrict__ fn, int F) {
  int f = blockIdx.x * blockDim.x + threadIdx.x;
  if (f >= F) return;
  int i0 = faces[f*3+0], i1 = faces[f*3+1], i2 = faces[f*3+2];
  float ax = verts[i0*3+0], ay = verts[i0*3+1], az = verts[i0*3+2];
  float bx = verts[i1*3+0], by = verts[i1*3+1], bz = verts[i1*3+2];
  float cx = verts[i2*3+0], cy = verts[i2*3+1], cz = verts[i2*3+2];
  float e1x = bx-ax, e1y = by-ay, e1z = bz-az;
  float e2x = cx-ax, e2y = cy-ay, e2z = cz-az;
  float nx = e1y*e2z - e1z*e2y;
  float ny = e1z*e2x - e1x*e2z;
  float nz = e1x*e2y - e1y*e2x;
  float l = fmaxf(sqrtf(nx*nx+ny*ny+nz*nz), 1e-12f);
  fn[f*3+0] = nx/l; fn[f*3+1] = ny/l; fn[f*3+2] = nz/l;
}

// ------- Kernel 2: vertex normals (deterministic per-vertex scan) -------
__global__ void k_vertex_normals(const int* __restrict__ faces,
                                 const float* __restrict__ fn,
                                 float* __restrict__ vn, int F, int V) {
  __shared__ int sf[512*3];
  int v = blockIdx.x * blockDim.x + threadIdx.x;
  float ax = 0.f, ay = 0.f, az = 0.f;
  for (int c0 = 0; c0 < F; c0 += 512) {
    int n = min(512, F - c0);
    __syncthreads();
    for (int i = threadIdx.x; i < n*3; i += blockDim.x) sf[i] = faces[c0*3 + i];
    __syncthreads();
    if (v < V) {
      for (int j = 0; j < n; ++j) {
        int m = (sf[j*3+0]==v) + (sf[j*3+1]==v) + (sf[j*3+2]==v);
        if (m) {
          float w = (float)m;
          ax += w*fn[(c0+j)*3+0]; ay += w*fn[(c0+j)*3+1]; az += w*fn[(c0+j)*3+2];
        }
      }
    }
  }
  if (v < V) {
    float l = fmaxf(sqrtf(ax*ax+ay*ay+az*az), 1e-12f);
    vn[v*3+0] = ax/l; vn[v*3+1] = ay/l; vn[v*3+2] = az/l;
  }
}

// select M4row[nn] without dynamic local indexing (keeps registers)
__device__ __forceinline__ float pick4(float c0, float c1, float c2, float c3, int nn) {
  float r01 = (nn & 1) ? c1 : c0;
  float r23 = (nn & 1) ? c3 : c2;
  float r   = (nn & 2) ? r23 : r01;
  return (nn < 4) ? r : 0.f;
}

// --------- Kernel 3: clip transform via V_WMMA_F32_16X16X4_F32 ---------
// One wave handles one 16-vertex tile: clip_tile(16x4) = A(16x4) x M4(4x4 in 4x16 B).
// A layout per ISA (16x4 f32): lanes 0-15 row M, v0=K0 v1=K1; lanes 16-31 row M, v0=K2 v1=K3.
// B assumed symmetric: lane = col N, lanes 0-15 hold K0/K1 in v0/v1, lanes 16-31 hold K2/K3.
__global__ void k_transform(const float* __restrict__ verts,
                            const float* __restrict__ pose,
                            ProjMat P, float* __restrict__ clip, int V) {
  int lane  = threadIdx.x & 31;
  int wave  = (blockIdx.x * blockDim.x + threadIdx.x) >> 5;
  int vbase = wave * 16;
  if (vbase >= V) return;   // wave-uniform

  // Mv[i][j] = cam_pose[j][i]  (view = cam_pose^T), cam_pose rows 0..2 sign-flipped by gl=diag(1,-1,-1)
  float Mv[4][4];
#pragma unroll
  for (int i = 0; i < 4; ++i) {
    Mv[i][0] =  pose[0*4+i];
    Mv[i][1] = -pose[1*4+i];
    Mv[i][2] = -pose[2*4+i];
    Mv[i][3] =  pose[3*4+i];
  }
  // M4[k][n] = sum_j Mv[k][j] * P[n][j]   (clip = obj @ Mv @ proj.T-as-stored)
  float M4[4][4];
#pragma unroll
  for (int k = 0; k < 4; ++k)
#pragma unroll
    for (int n = 0; n < 4; ++n) {
      float s = 0.f;
#pragma unroll
      for (int j = 0; j < 4; ++j) s += Mv[k][j] * P.p[n*4+j];
      M4[k][n] = s;
    }

  // A operand
  int r = lane & 15;
  int v = vbase + r; if (v >= V) v = V - 1;
  v2f a;
  if (lane < 16) { a.x = verts[v*3+0]; a.y = verts[v*3+1]; }
  else           { a.x = verts[v*3+2]; a.y = 1.0f; }

  // B operand (zero-padded beyond column 3)
  int n = lane & 15;
  float b0lo = pick4(M4[0][0], M4[0][1], M4[0][2], M4[0][3], n);
  float b1lo = pick4(M4[1][0], M4[1][1], M4[1][2], M4[1][3], n);
  float b0hi = pick4(M4[2][0], M4[2][1], M4[2][2], M4[2][3], n);
  float b1hi = pick4(M4[3][0], M4[3][1], M4[3][2], M4[3][3], n);
  v2f b;
  b.x = (lane < 16) ? b0lo : b0hi;
  b.y = (lane < 16) ? b1lo : b1hi;

  v8f c = {};
  // 8 args: (neg_a, A, neg_b, B, c_mod, C, reuse_a, reuse_b)
  v8f d = __builtin_amdgcn_wmma_f32_16x16x4_f32(false, a, false, b, (short)0, c, false, false);

  union { v8f vv; float f[8]; } u; u.vv = d;
  // D layout 16x16 f32: vgpr r, lanes 0-15 -> row r, col=lane; lanes 16-31 -> row r+8, col=lane-16
  int col = lane & 15;
  if (col < 4) {
    int rb = (lane < 16) ? 0 : 8;
#pragma unroll
    for (int rr = 0; rr < 8; ++rr) {
      int vo = vbase + rb + rr;
      if (vo < V) clip[vo*4 + col] = u.f[rr];
    }
  }
}

// --------- Kernel 4: per-vertex 11-channel colors ---------
__global__ void k_vertex_colors(const float* __restrict__ verts,
                                const float* __restrict__ pose,
                                const float* __restrict__ vn,
                                float* __restrict__ colors, int V) {
  int v = blockIdx.x * blockDim.x + threadIdx.x;
  if (v >= V) return;
  float x = verts[v*3+0], y = verts[v*3+1], z = verts[v*3+2];
  float* cc = colors + (size_t)v*11;
  cc[0] = 1.0f;
#pragma unroll
  for (int j = 0; j < 3; ++j)   // vert_3d = obj3 @ pose[:3,:3]^T + t
    cc[1+j] = pose[j*4+0]*x + pose[j*4+1]*y + pose[j*4+2]*z + pose[j*4+3];
  cc[4] = vn[v*3+0]; cc[5] = vn[v*3+1]; cc[6] = vn[v*3+2];
  cc[7] = x; cc[8] = y; cc[9] = z; cc[10] = 1.0f;
}

// --------- Kernel 5: per-face raster record (16 floats) ---------
// [ax,ay,bx,by,cx,cy, z0,z1,z2, iw0,iw1,iw2, inv_area, valid, 0, 0]
__global__ void k_face_setup(const float* __restrict__ clip,
                             const int* __restrict__ faces,
                             float* __restrict__ fd, int F) {
  int f = blockIdx.x * blockDim.x + threadIdx.x;
  if (f >= F) return;
  float px[3], py[3], pz[3], iw[3];
  bool wok = true;
#pragma unroll
  for (int k = 0; k < 3; ++k) {
    int v = faces[f*3+k];
    float cx = clip[v*4+0], cy = clip[v*4+1], cz = clip[v*4+2], w = clip[v*4+3];
    float ww = (fabsf(w) > 1e-9f) ? w : 1e-9f;
    float invw = 1.0f / ww;
    px[k] = (cx*invw + 1.0f) * 0.5f * (float)IMG_W;
    py[k] = (1.0f - cy*invw) * 0.5f * (float)IMG_H;
    pz[k] = cz*invw;
    iw[k] = invw;
    wok = wok && (w > 1e-6f);
  }
  float area = (px[1]-px[0])*(py[2]-py[0]) - (py[1]-py[0])*(px[2]-px[0]);
  bool aok = fabsf(area) > 1e-12f;
  float inva = 1.0f / (aok ? area : 1e-12f);
  float* t = fd + (size_t)f*16;
  t[0]=px[0]; t[1]=py[0]; t[2]=px[1]; t[3]=py[1]; t[4]=px[2]; t[5]=py[2];
  t[6]=pz[0]; t[7]=pz[1]; t[8]=pz[2];
  t[9]=iw[0]; t[10]=iw[1]; t[11]=iw[2];
  t[12]=inva; t[13]=(wok && aok) ? 1.0f : 0.0f; t[14]=0.0f; t[15]=0.0f;
}

// --------- Kernel 6: per-pixel raster + perspective-correct shade ---------
__global__ void k_raster(const float* __restrict__ fd,
                         const int* __restrict__ faces,
                         const float* __restrict__ colors,
                         float* __restrict__ out, int F) {
  __shared__ float lds[128*16];
  int pix = blockIdx.x * blockDim.x + threadIdx.x;
  int xi = pix % IMG_W, yi = pix / IMG_W;
  float X = (float)xi + 0.5f, Y = (float)yi + 0.5f;

  float bestD = 1e9f;           // BIG, strict < => earliest face wins ties (matches chunked argmin)
  int   bestF = -1;
  float bb0 = 0.f, bb1 = 0.f;

  for (int c0 = 0; c0 < F; c0 += 128) {
    int nf = min(128, F - c0);
    __syncthreads();
    for (int i = threadIdx.x; i < nf*16; i += blockDim.x) lds[i] = fd[(size_t)c0*16 + i];
    __syncthreads();
    for (int j = 0; j < nf; ++j) {
      const float* t = &lds[j*16];
      float valid = t[13];
      if (valid == 0.f) continue;
      float ax=t[0], ay=t[1], bx=t[2], by=t[3], cx=t[4], cy=t[5];
      float inva = t[12];
      float e0 = ((X-bx)*(cy-by) - (Y-by)*(cx-bx)) * inva;
      float e1 = ((X-cx)*(ay-cy) - (Y-cy)*(ax-cx)) * inva;
      float e2 = 1.0f - e0 - e1;
      float depth = e0*t[6] + e1*t[7] + e2*t[8];
      bool inside = (e0 >= 0.f) & (e1 >= 0.f) & (e2 >= 0.f) &
                    (depth >= -1.0f) & (depth <= 1.0f);
      if (inside && depth < bestD) { bestD = depth; bestF = c0 + j; bb0 = e0; bb1 = e1; }
    }
  }

  if (pix < IMG_H*IMG_W) {
    float oc[11];
    if (bestF >= 0) {
      float b2 = 1.0f - bb0 - bb1;
      const float* t = fd + (size_t)bestF*16;
      float p0 = bb0*t[9], p1 = bb1*t[10], p2 = b2*t[11];
      float den = p0 + p1 + p2;
      den = (fabsf(den) > 1e-12f) ? den : 1e-12f;
      float rden = 1.0f/den;
      int i0 = faces[bestF*3+0], i1 = faces[bestF*3+1], i2 = faces[bestF*3+2];
      const float* ca = colors + (size_t)i0*11;
      const float* cb = colors + (size_t)i1*11;
      const float* cg = colors + (size_t)i2*11;
#pragma unroll
      for (int cch = 0; cch < 11; ++cch)
        oc[cch] = (p0*ca[cch] + p1*cb[cch] + p2*cg[cch]) * rden;
    } else {
#pragma unroll
      for (int cch = 0; cch < 11; ++cch) oc[cch] = 0.f;
    }
#pragma unroll
    for (int cch = 0; cch < 11; ++cch) out[(size_t)pix*11 + cch] = oc[cch];
  }
}

extern "C" void kernel_launch(void* const* d_in, const int* in_sizes, int n_in,
                              void* d_out, int out_size, void* d_ws, size_t ws_size,
                              hipStream_t stream) {
  const float* verts = (const float*)d_in[0];   // (1,V,3) f32
  const int*   faces = (const int*)d_in[1];     // (1,F,3) i32
  const float* pose  = (const float*)d_in[2];   // (1,4,4) f32
  int V = in_sizes[0] / 3;
  int F = in_sizes[1] / 3;

  // workspace carve (floats)
  float* ws = (float*)d_ws;
  size_t o = 0;
  float* fn     = ws + o; o += (size_t)F*3;
  float* vn     = ws + o; o += (size_t)V*3;
  float* clip   = ws + o; o += (size_t)V*4;
  float* colors = ws + o; o += (size_t)V*11;
  float* fd     = ws + o; o += (size_t)F*16;
  float* out = (float*)d_out;

  // projection matrix built in double, matching build_projection (x0=y0=0)
  const double fx = 286.2057, fy = 286.7852, cxd = 162.6306, cyd = 121.0245;
  const double Wd = IMG_W, Hd = IMG_H, nc = 0.1, fc = 10.0;
  const double q  = -(fc + nc) / (fc - nc);
  const double qn = -2.0 * fc * nc / (fc - nc);
  double pm[16] = { 2.0*fx/Wd, 0.0,        (-2.0*cxd + Wd)/Wd, 0.0,
                    0.0,       2.0*fy/Hd,  ( 2.0*cyd - Hd)/Hd, 0.0,   // row1 already *(-1)
                    0.0,       0.0,        q,                  qn,
                    0.0,       0.0,       -1.0,                0.0 };
  ProjMat P;
  for (int i = 0; i < 16; ++i) P.p[i] = (float)pm[i];

  dim3 blk(256);
  k_face_normals<<<(F + 255)/256, blk, 0, stream>>>(verts, faces, fn, F);
  k_vertex_normals<<<(V + 255)/256, blk, 0, stream>>>(faces, fn, vn, F, V);

  int tiles = (V + 15) / 16;               // one wave per tile, 8 waves/block
  int tblocks = (tiles + 7) / 8;
  k_transform<<<tblocks, blk, 0, stream>>>(verts, pose, P, clip, V);

  k_vertex_colors<<<(V + 255)/256, blk, 0, stream>>>(verts, pose, vn, colors, V);
  k_face_setup<<<(F + 255)/256, blk, 0, stream>>>(clip, faces, fd, F);

  int npix = IMG_H * IMG_W;
  k_raster<<<(npix + 255)/256, blk, 0, stream>>>(fd, faces, colors, out, F);
}